// NCEAverage_41755672051818
// MI455X (gfx1250) — compile-verified
//
#include <hip/hip_runtime.h>
#include <math.h>

// NCEAverage for MI455X (gfx1250), wave32.
// Memory-bound problem (~1.6 GB traffic, 0.27 GFLOP): bulk NT copy of both
// memory banks + WMMA f32 16x16x4 batched dot-product scoring + tiny EMA.

typedef __attribute__((ext_vector_type(2))) float v2f;
typedef __attribute__((ext_vector_type(4))) float f4;
typedef __attribute__((ext_vector_type(8))) float v8f;

#define BATCH   256
#define DIM     128
#define KP1     2049          // K+1
#define NROWS   500000
#define NTILES  129           // ceil(2049/16)
#define INV_T   (1.0f / 0.07f)

// ---------------------------------------------------------------------------
// Kernel 1: stream both memory banks to the output regions (non-temporal).
// ---------------------------------------------------------------------------
__global__ __launch_bounds__(256) void bank_copy(const f4* __restrict__ src0,
                                                 const f4* __restrict__ src1,
                                                 f4* __restrict__ dst0,
                                                 f4* __restrict__ dst1,
                                                 int n4) {
  int i = blockIdx.x * blockDim.x + threadIdx.x;
  if (i < n4) {
    f4 a = __builtin_nontemporal_load(src0 + i);
    __builtin_nontemporal_store(a, dst0 + i);
    f4 b = __builtin_nontemporal_load(src1 + i);
    __builtin_nontemporal_store(b, dst1 + i);
  }
}

// ---------------------------------------------------------------------------
// Kernel 2: WMMA scoring.  One wave per (16-k tile, sample b, bank).
// A fragment (16x4 f32): lane l holds pair row[4c + 2*(l>>4)] of gathered row
//   idx[b, k0 + (l&15)]  -> a single global b64 load per lane IS the fragment.
// B fragment: every lane gets the same x pair -> all 16 D-columns compute the
//   identical (correct) dot product; lanes 0 and 16 hold the 16 results.
// ---------------------------------------------------------------------------
__global__ __launch_bounds__(128) void score_wmma(
    const float* __restrict__ cdr, const float* __restrict__ vdj,
    const int* __restrict__ idx,
    const float* __restrict__ mem_cdr, const float* __restrict__ mem_vdj,
    float* __restrict__ out) {
  const int lane = threadIdx.x & 31;
  const int wave = threadIdx.x >> 5;
  const int t = blockIdx.x * 4 + wave;      // k-tile index
  if (t >= NTILES) return;                  // wave-uniform exit (EXEC all-1 at WMMA)
  const int b    = blockIdx.y;
  const int bank = blockIdx.z;              // 0: out_cdr = <mem_vdj, cdr>, 1: out_vdj = <mem_cdr, vdj>
  const float* __restrict__ mem = bank ? mem_cdr : mem_vdj;
  const float* __restrict__ x   = (bank ? vdj : cdr) + (size_t)b * DIM;
  float* __restrict__ outp = out + (size_t)bank * (BATCH * KP1) + (size_t)b * KP1;

  const int m  = lane & 15;                 // row of tile (M)
  const int hi = lane >> 4;                 // which K half-pair
  int k = t * 16 + m;
  if (k >= KP1) k = KP1 - 1;                // clamp (last tile has 1 valid row)
  const long long row = (long long)idx[b * KP1 + k];
  const float* __restrict__ rp = mem + (size_t)row * DIM;

  v8f acc = {0.f, 0.f, 0.f, 0.f, 0.f, 0.f, 0.f, 0.f};
#pragma unroll
  for (int c = 0; c < 32; ++c) {            // D=128 in chunks of K=4
    v2f a  = __builtin_nontemporal_load((const v2f*)(rp + 4 * c + 2 * hi));
    v2f bx = *(const v2f*)(x + 4 * c + 2 * hi);
    acc = __builtin_amdgcn_wmma_f32_16x16x4_f32(
        /*neg_a=*/false, a, /*neg_b=*/false, bx,
        /*c_mod=*/(short)0, acc, /*reuse_a=*/false, /*reuse_b=*/false);
  }

  // D layout: VGPR r -> M=r (lanes 0-15), M=8+r (lanes 16-31); all N identical.
  if (m == 0) {
    const int kbase = t * 16 + hi * 8;
#pragma unroll
    for (int r = 0; r < 8; ++r) {
      const int kk = kbase + r;
      if (kk < KP1) outp[kk] = acc[r] * INV_T;
    }
  }
}

// ---------------------------------------------------------------------------
// Kernel 3: EMA + L2-normalize 256 rows per bank, scatter into copied banks.
// One wave per (sample, bank). Deterministic duplicate handling: highest b wins.
// ---------------------------------------------------------------------------
__global__ __launch_bounds__(32) void ema_update(
    const float* __restrict__ cdr, const float* __restrict__ vdj,
    const int* __restrict__ indexp,
    const float* __restrict__ mem_cdr, const float* __restrict__ mem_vdj,
    float* __restrict__ out_mem_cdr, float* __restrict__ out_mem_vdj) {
  const int b    = blockIdx.x;
  const int bank = blockIdx.y;              // 0: cdr bank, 1: vdj bank
  const int pos  = indexp[b];
  for (int j = b + 1; j < BATCH; ++j)
    if (indexp[j] == pos) return;           // let the last duplicate win

  const float* __restrict__ mem = (bank ? mem_vdj : mem_cdr) + (size_t)pos * DIM;
  const float* __restrict__ x   = (bank ? vdj : cdr) + (size_t)b * DIM;
  float* __restrict__ dst = (bank ? out_mem_vdj : out_mem_cdr) + (size_t)pos * DIM;

  const int lane = threadIdx.x;             // 32 lanes x 4 floats = 128
  f4 mv = *(const f4*)(mem + lane * 4);
  f4 xv = *(const f4*)(x + lane * 4);
  f4 p  = 0.5f * mv + 0.5f * xv;            // MOM = 0.5
  float s = p.x * p.x + p.y * p.y + p.z * p.z + p.w * p.w;
#pragma unroll
  for (int off = 16; off >= 1; off >>= 1)
    s += __shfl_xor(s, off, 32);
  const float inv = 1.0f / sqrtf(s);
  f4 r = p * inv;
  *(f4*)(dst + lane * 4) = r;
}

// ---------------------------------------------------------------------------
extern "C" void kernel_launch(void* const* d_in, const int* in_sizes, int n_in,
                              void* d_out, int out_size, void* d_ws, size_t ws_size,
                              hipStream_t stream) {
  const float* cdr     = (const float*)d_in[0];
  const float* vdj     = (const float*)d_in[1];
  const int*   index   = (const int*)d_in[2];
  const int*   idx     = (const int*)d_in[3];
  const float* mem_cdr = (const float*)d_in[4];
  const float* mem_vdj = (const float*)d_in[5];

  float* out          = (float*)d_out;
  float* out_mem_cdr  = out + 2 * (size_t)BATCH * KP1;          // after out_cdr, out_vdj
  float* out_mem_vdj  = out_mem_cdr + (size_t)NROWS * DIM;

  // 1) stream both banks to output (1 GB of the ~1.6 GB roofline)
  const int n4 = NROWS * DIM / 4;                               // 16,000,000 float4 per bank
  bank_copy<<<(n4 + 255) / 256, 256, 0, stream>>>(
      (const f4*)mem_cdr, (const f4*)mem_vdj,
      (f4*)out_mem_cdr, (f4*)out_mem_vdj, n4);

  // 2) WMMA scoring: grid = (tile groups of 4 waves, B, 2 banks)
  dim3 gs((NTILES + 3) / 4, BATCH, 2);                          // 33 x 256 x 2
  score_wmma<<<gs, 128, 0, stream>>>(cdr, vdj, idx, mem_cdr, mem_vdj, out);

  // 3) EMA rows (after copy; stream order guarantees this)
  dim3 ge(BATCH, 2);
  ema_update<<<ge, 32, 0, stream>>>(cdr, vdj, index, mem_cdr, mem_vdj,
                                    out_mem_cdr, out_mem_vdj);
}